// DynamicConv1d_68101001445546
// MI455X (gfx1250) — compile-verified
//
#include <hip/hip_runtime.h>

// DynamicConv1d fused: logits GEMM (FP32 WMMA) + softmax(K=7) + causal depthwise conv
// T=2048, B=4, C=1024, H=16, K=7, R=64, O=H*K=112, M=T*B=8192

#define T_DIM 2048
#define B_DIM 4
#define C_DIM 1024
#define H_DIM 16
#define K_DIM 7
#define O_DIM 112

typedef __attribute__((ext_vector_type(2))) float v2f;
typedef __attribute__((ext_vector_type(8))) float v8f;

__global__ __launch_bounds__(256)
void dynconv_fused_kernel(const float* __restrict__ x,
                          const float* __restrict__ Wm,
                          const float* __restrict__ bias,
                          float* __restrict__ out) {
  __shared__ float s_w[16 * O_DIM];   // 16 rows x 112 logits -> softmaxed weights

  const int mBase = blockIdx.x << 4;  // 16 rows of the (T*B, C) problem
  const int tid   = threadIdx.x;
  const int wave  = tid >> 5;
  const int lane  = tid & 31;
  const int half  = lane >> 4;        // 0: lanes 0-15, 1: lanes 16-31
  const int lm    = lane & 15;

  // ---------- Phase 1: logits[16x112] = x_tile(16x1024) * W^T(1024x112) ----------
  // V_WMMA_F32_16X16X4_F32:  A 16x4 (2 VGPR: lane<16 holds K=0,1; lane>=16 holds K=2,3)
  //                          B 4x16 (same K-striping), C/D 16x16 in 8 VGPRs.
  if (wave < 7) {
    const int oBase = wave << 4;
    const float* __restrict__ xr = x  + (size_t)(mBase + lm) * C_DIM + 2 * half;
    const float* __restrict__ wr = Wm + (size_t)(oBase + lm) * C_DIM + 2 * half;
    v8f acc = {0.f, 0.f, 0.f, 0.f, 0.f, 0.f, 0.f, 0.f};
    #pragma unroll 4
    for (int k = 0; k < C_DIM; k += 4) {
      v2f a = *(const v2f*)(xr + k);   // A[m=lm][k0 + 2*half + {0,1}]
      v2f b = *(const v2f*)(wr + k);   // B[k0 + 2*half + {0,1}][n=lm] = W[oBase+lm][..]
      acc = __builtin_amdgcn_wmma_f32_16x16x4_f32(false, a, false, b,
                                                  (short)0, acc, false, false);
    }
    // C/D layout: VGPR v holds row M = v + 8*half, column N = lm (within tile)
    #pragma unroll
    for (int v = 0; v < 8; ++v)
      s_w[(v + 8 * half) * O_DIM + oBase + lm] = acc[v];
  }
  __syncthreads();

  // ---------- Phase 2: softmax over K=7, one (row, head) group per thread ----------
  {
    const int row = tid >> 4;   // 0..15
    const int h   = tid & 15;   // 0..15
    float* g = s_w + row * O_DIM + h * K_DIM;
    float mx = g[0];
    #pragma unroll
    for (int k = 1; k < K_DIM; ++k) mx = fmaxf(mx, g[k]);
    float e[K_DIM];
    float s = 0.f;
    #pragma unroll
    for (int k = 0; k < K_DIM; ++k) { e[k] = __expf(g[k] - mx); s += e[k]; }
    const float inv = 1.0f / s;
    #pragma unroll
    for (int k = 0; k < K_DIM; ++k) g[k] = e[k] * inv;
  }
  __syncthreads();

  // ---------- Phase 3: causal dynamic conv for the 16 rows, all 1024 channels ----------
  // out[m, c] = bias[c] + sum_k x[m - (6-k)*B, c] * w[row, c/64, k]   (t-6+k >= 0)
  for (int i = tid; i < 16 * C_DIM; i += 256) {
    const int row = i >> 10;
    const int c   = i & (C_DIM - 1);
    const int m   = mBase + row;
    const int t   = m >> 2;                          // B == 4
    const float* wv = s_w + row * O_DIM + (c >> 6) * K_DIM;
    float accv = bias[c];
    if (t >= K_DIM - 1) {
      #pragma unroll
      for (int k = 0; k < K_DIM; ++k)
        accv = fmaf(x[(size_t)(m - (K_DIM - 1 - k) * B_DIM) * C_DIM + c], wv[k], accv);
    } else {
      #pragma unroll
      for (int k = 0; k < K_DIM; ++k) {
        if (t - (K_DIM - 1) + k >= 0)
          accv = fmaf(x[(size_t)(m - (K_DIM - 1 - k) * B_DIM) * C_DIM + c], wv[k], accv);
      }
    }
    out[(size_t)m * C_DIM + c] = accv;
  }
}

extern "C" void kernel_launch(void* const* d_in, const int* in_sizes, int n_in,
                              void* d_out, int out_size, void* d_ws, size_t ws_size,
                              hipStream_t stream) {
  const float* x    = (const float*)d_in[0];   // (T, B, C) f32
  const float* Wm   = (const float*)d_in[1];   // (H*K, C)  f32
  const float* bias = (const float*)d_in[2];   // (C,)      f32
  float* out = (float*)d_out;                  // (T, B, C) f32

  const int blocks = (T_DIM * B_DIM) / 16;     // 512 blocks, 256 threads (8 waves)
  hipLaunchKernelGGL(dynconv_fused_kernel, dim3(blocks), dim3(256), 0, stream,
                     x, Wm, bias, out);
}